// ExtendP_65566970741223
// MI455X (gfx1250) — compile-verified
//
#include <hip/hip_runtime.h>

typedef __attribute__((ext_vector_type(2))) float v2f;
typedef __attribute__((ext_vector_type(8))) float v8f;

// Problem constants (from reference): B=32, L=64, N=10, C=2, G=32
constexpr unsigned kSlab     = 10u * 10u * 2u * 2u * 32u;   // 12800 floats per (b,l)
constexpr unsigned kRegion   = 63u * kSlab;                 // 806400: sliced floats per batch
constexpr unsigned kFull     = 64u * kSlab;                 // 819200: full floats per batch
constexpr unsigned kCsFloats = 32u * kRegion;               // 25,804,800 floats to sum per cs tensor
constexpr unsigned kChunk    = 128u;                        // floats per wave-iteration
constexpr unsigned kNumChunks = kCsFloats / kChunk;         // 201,600 (exact; kRegion % 128 == 0)
constexpr unsigned kTp       = 10u * 10u * 2u * 32u;        // 6400 floats per trans_p tensor

// Stage 1: bandwidth-bound streaming sum of cs_mu[:, :-1] + cs_var[:, :-1].
// Each wave iteration: 2x global_load_b128 (coalesced 512B/wave/array),
// packed f32 add, then 2x v_wmma_f32_16x16x4_f32 with B = ones -> one-instruction
// wave-wide 64-float row-sum accumulated in f32.
__global__ __launch_bounds__(256) void cs_reduce_wmma(const float* __restrict__ mu,
                                                      const float* __restrict__ var,
                                                      float* __restrict__ partial) {
    const unsigned lane  = threadIdx.x & 31u;
    const unsigned wave  = threadIdx.x >> 5;
    const unsigned gwave = blockIdx.x * 8u + wave;
    const unsigned nW    = gridDim.x * 8u;

    v8f acc = {};
    const v2f ones = {1.0f, 1.0f};

    for (unsigned c = gwave; c < kNumChunks; c += nW) {
        // Map flat sliced index -> memory index (skip each batch's last L slab).
        const unsigned f0   = c * kChunk;
        const unsigned b    = f0 / kRegion;          // const divide -> mul/shift
        const unsigned r    = f0 - b * kRegion;
        const unsigned base = b * kFull + r + lane * 4u;   // 16B-aligned per lane

        const float4 x = *reinterpret_cast<const float4*>(mu  + base);
        const float4 y = *reinterpret_cast<const float4*>(var + base);

        const v2f a0 = {x.x + y.x, x.y + y.y};
        const v2f a1 = {x.z + y.z, x.w + y.w};

        // D[m,n] += sum_k A[m,k] : matrix engine as wave-wide reduction unit.
        acc = __builtin_amdgcn_wmma_f32_16x16x4_f32(false, a0, false, ones,
                                                    (short)0, acc, false, false);
        acc = __builtin_amdgcn_wmma_f32_16x16x4_f32(false, a1, false, ones,
                                                    (short)0, acc, false, false);
    }

    // Every column of D is identical (B was all-ones). Per-lane: sum the 8 acc
    // VGPRs. Lane 0 holds M=0..7 partial of column 0; lane 16 holds M=8..15.
    float s = acc[0] + acc[1] + acc[2] + acc[3] + acc[4] + acc[5] + acc[6] + acc[7];
    const float wave_total = __shfl(s, 0, 32) + __shfl(s, 16, 32);

    __shared__ float lds[8];
    if (lane == 0) lds[wave] = wave_total;
    __syncthreads();
    if (threadIdx.x == 0) {
        float t = 0.0f;
        #pragma unroll
        for (int i = 0; i < 8; ++i) t += lds[i];
        partial[blockIdx.x] = t;   // fixed order -> deterministic
    }
}

// Stage 2: single block. Deterministic fixed-order reduction of block partials
// and of the tiny trans_p tensors; apply broadcast-replication weights.
__global__ __launch_bounds__(256) void finalize_sum(const float* __restrict__ partial,
                                                    unsigned nPartial,
                                                    const float* __restrict__ tpmu,
                                                    const float* __restrict__ tpvar,
                                                    float* __restrict__ out) {
    __shared__ float s_cs[256];
    __shared__ float s_tp[256];
    const unsigned t = threadIdx.x;

    float cs = 0.0f;
    for (unsigned i = t; i < nPartial; i += 256u) cs += partial[i];
    float tp = 0.0f;
    for (unsigned i = t; i < kTp; i += 256u) tp += tpmu[i] + tpvar[i];

    s_cs[t] = cs;
    s_tp[t] = tp;
    __syncthreads();
    for (unsigned stride = 128u; stride > 0u; stride >>= 1) {
        if (t < stride) {
            s_cs[t] += s_cs[t + stride];
            s_tp[t] += s_tp[t + stride];
        }
        __syncthreads();
    }
    if (t == 0) {
        // cs terms replicated over one N and one C dim: *20
        // trans_p terms replicated over B,(L-1),N,C,C: *80640
        out[0] = s_cs[0] * 20.0f + s_tp[0] * 80640.0f;
    }
}

extern "C" void kernel_launch(void* const* d_in, const int* in_sizes, int n_in,
                              void* d_out, int out_size, void* d_ws, size_t ws_size,
                              hipStream_t stream) {
    const float* cs_mu  = (const float*)d_in[0];
    const float* cs_var = (const float*)d_in[1];
    const float* tpmu   = (const float*)d_in[2];
    const float* tpvar  = (const float*)d_in[3];
    float* out     = (float*)d_out;
    float* partial = (float*)d_ws;

    unsigned nBlocks = 1024u;                       // 8192 waves: saturates 23.3 TB/s HBM
    const unsigned maxBlocks = (unsigned)(ws_size / sizeof(float));
    if (nBlocks > maxBlocks) nBlocks = maxBlocks;   // fixed ws_size -> still deterministic
    if (nBlocks == 0u) nBlocks = 1u;

    cs_reduce_wmma<<<nBlocks, 256, 0, stream>>>(cs_mu, cs_var, partial);
    finalize_sum<<<1, 256, 0, stream>>>(partial, nBlocks, tpmu, tpvar, out);
}